// CNFBlock_42614665511220
// MI455X (gfx1250) — compile-verified
//
#include <hip/hip_runtime.h>

typedef __attribute__((ext_vector_type(2))) float v2f;
typedef __attribute__((ext_vector_type(4))) float v4f;
typedef __attribute__((ext_vector_type(8))) float v8f;

#define D_DIM 256
#define ZERO8 {0.f,0.f,0.f,0.f,0.f,0.f,0.f,0.f}

__device__ __forceinline__ v8f wmma4(v2f a, v2f b, v8f c) {
  // V_WMMA_F32_16X16X4_F32: D = A(16x4,f32) * B(4x16,f32) + C(16x16,f32)
  return __builtin_amdgcn_wmma_f32_16x16x4_f32(
      /*neg_a=*/false, a, /*neg_b=*/false, b,
      /*c_mod=*/(short)0, c, /*reuse_a=*/false, /*reuse_b=*/false);
}

__device__ __forceinline__ v2f ld2(const float* p) { return *(const v2f*)p; }

// ---------------------------------------------------------------------------
// Kernel H: ct[t] = -0.5*||h[t]||^2 - (D/2)*log(2*pi)
// ---------------------------------------------------------------------------
__global__ void hsq_kernel(const float* __restrict__ h, float* __restrict__ ct, int T) {
  int t = blockIdx.x * blockDim.x + threadIdx.x;
  if (t >= T) return;
  const float* row = h + (size_t)t * D_DIM;
  float s = 0.f;
  #pragma unroll 4
  for (int k = 0; k < D_DIM; k += 4) {
    v4f x = *(const v4f*)(row + k);
    s += x.x * x.x + x.y * x.y + x.z * x.z + x.w * x.w;
  }
  ct[t] = -0.5f * s - 128.0f * 1.8378770664093453f; // -(D/2)*log(2*pi), D=256
}

// ---------------------------------------------------------------------------
// Kernel A: CNF RK4 integration over emb rows; writes cv[v] = -0.5*e_sq - dl.
// One workgroup = 16 vocab rows, 8 waves; wave w owns column tiles 2w, 2w+1.
// z tile kept in LDS for all 8*4 stage-GEMMs; Wx streamed from L2.
// ---------------------------------------------------------------------------
__global__ __launch_bounds__(256) void cnf_kernel(
    const float* __restrict__ emb, const float* __restrict__ Wx,
    const float* __restrict__ wt,  const float* __restrict__ bvec,
    float* __restrict__ cv, int V) {
  __shared__ float zcur[16 * D_DIM];
  __shared__ float ztmp[16 * D_DIM];
  __shared__ float zacc[16 * D_DIM];
  __shared__ float esqS[16];
  __shared__ float dlS[16];

  const int tid  = threadIdx.x;
  const int v0   = blockIdx.x * 16;
  const int wave = tid >> 5;
  const int lane = tid & 31;
  const int half = lane >> 4;
  const int l    = lane & 15;
  const float dtC = 1.0f / 8.0f;

  if (tid < 16) { esqS[tid] = 0.f; dlS[tid] = 0.f; }
  // load z tile (= emb rows) and zero zacc; fuse e_sq partials
  {
    const int row = tid >> 4;   // 0..15
    const int seg = tid & 15;   // 0..15 -> 16 contiguous cols
    float s = 0.f;
    #pragma unroll
    for (int c = 0; c < 16; ++c) {
      int col = seg * 16 + c;
      float e = emb[(size_t)(v0 + row) * D_DIM + col];
      zcur[row * D_DIM + col] = e;
      zacc[row * D_DIM + col] = 0.f;
      s += e * e;
    }
    __syncthreads();
    atomicAdd(&esqS[row], s);
  }

  // per-wave loop-invariant column data (two 16-wide column tiles per wave)
  const int i0 = (wave * 2 + 0) * 16 + l;
  const int i1 = (wave * 2 + 1) * 16 + l;
  const float* WxRow0 = Wx + (size_t)i0 * D_DIM;
  const float* WxRow1 = Wx + (size_t)i1 * D_DIM;
  const float dW0 = Wx[(size_t)i0 * D_DIM + i0];
  const float dW1 = Wx[(size_t)i1 * D_DIM + i1];
  const float wt0 = wt[i0], wt1 = wt[i1];
  const float bv0 = bvec[i0], bv1 = bvec[i1];
  __syncthreads();

  // one RK4 stage: pre = zin @ Wx^T + tt*wt + b (via WMMA), ReLU, divergence
  auto stage = [&](const float* zin, float tt, float w, float cnext, bool hasNext) {
    v8f acc0 = ZERO8, acc1 = ZERO8;
    const float* aBase = zin + (size_t)l * D_DIM + 2 * half;
    #pragma unroll 4
    for (int k = 0; k < D_DIM; k += 4) {
      v2f a  = ld2(aBase + k);                  // A from LDS (ds_load_b64)
      v2f b0 = ld2(WxRow0 + k + 2 * half);      // B from L2-resident Wx
      v2f b1 = ld2(WxRow1 + k + 2 * half);
      acc0 = wmma4(a, b0, acc0);
      acc1 = wmma4(a, b1, acc1);
    }
    __syncthreads();   // all waves done reading zin before ztmp is rewritten
    const float add0 = tt * wt0 + bv0;
    const float add1 = tt * wt1 + bv1;
    const float dlw  = -(dtC / 6.f) * w;
    #pragma unroll
    for (int r = 0; r < 8; ++r) {
      const int m = r + 8 * half;
      float pre0 = acc0[r] + add0;
      float pre1 = acc1[r] + add1;
      float dz0 = fmaxf(pre0, 0.f);
      float dz1 = fmaxf(pre1, 0.f);
      float dc = (pre0 > 0.f ? dW0 : 0.f) + (pre1 > 0.f ? dW1 : 0.f);
      if (dc != 0.f) atomicAdd(&dlS[m], dlw * dc);   // ds_add_f32
      zacc[m * D_DIM + i0] += w * dz0;               // lane-exclusive element
      zacc[m * D_DIM + i1] += w * dz1;
      if (hasNext) {
        ztmp[m * D_DIM + i0] = zcur[m * D_DIM + i0] + cnext * dtC * dz0;
        ztmp[m * D_DIM + i1] = zcur[m * D_DIM + i1] + cnext * dtC * dz1;
      }
    }
    __syncthreads();
  };

  for (int step = 0; step < 8; ++step) {
    const float t0 = (float)step * dtC;
    stage(zcur, t0,              1.f, 0.5f, true);   // k1
    stage(ztmp, t0 + 0.5f * dtC, 2.f, 0.5f, true);   // k2
    stage(ztmp, t0 + 0.5f * dtC, 2.f, 1.0f, true);   // k3
    stage(ztmp, t0 + dtC,        1.f, 0.0f, false);  // k4
    // z += dt/6 * (k1 + 2k2 + 2k3 + k4); reset accumulator
    #pragma unroll
    for (int r = 0; r < 8; ++r) {
      const int m = r + 8 * half;
      zcur[m * D_DIM + i0] += (dtC / 6.f) * zacc[m * D_DIM + i0];
      zacc[m * D_DIM + i0] = 0.f;
      zcur[m * D_DIM + i1] += (dtC / 6.f) * zacc[m * D_DIM + i1];
      zacc[m * D_DIM + i1] = 0.f;
    }
    __syncthreads();
  }

  if (tid < 16) cv[v0 + tid] = -0.5f * esqS[tid] - dlS[tid];
}

// ---------------------------------------------------------------------------
// Kernel B: out[t,v] = (hf @ emb^T)[t,v] + cv[v] + ct[t]
// One wave per 16x64 output tile; 4 WMMA accumulators share one A load.
// emb (32.8 MB) and hf (2 MB) are L2-resident -> output-bandwidth bound.
// ---------------------------------------------------------------------------
__global__ __launch_bounds__(32) void logp_kernel(
    const float* __restrict__ hf, const float* __restrict__ emb,
    const float* __restrict__ cv, const float* __restrict__ ct,
    float* __restrict__ out, int V) {
  const int lane = threadIdx.x & 31;
  const int half = lane >> 4;
  const int l    = lane & 15;
  const int col0 = blockIdx.x * 64;
  const int row0 = blockIdx.y * 16;

  const float* Arow = hf  + (size_t)(row0 + l) * D_DIM + 2 * half;
  const float* B0   = emb + (size_t)(col0 +  0 + l) * D_DIM + 2 * half;
  const float* B1   = emb + (size_t)(col0 + 16 + l) * D_DIM + 2 * half;
  const float* B2   = emb + (size_t)(col0 + 32 + l) * D_DIM + 2 * half;
  const float* B3   = emb + (size_t)(col0 + 48 + l) * D_DIM + 2 * half;

  v8f c0 = ZERO8, c1 = ZERO8, c2 = ZERO8, c3 = ZERO8;
  #pragma unroll 4
  for (int k = 0; k < D_DIM; k += 4) {
    v2f a = ld2(Arow + k);
    c0 = wmma4(a, ld2(B0 + k), c0);
    c1 = wmma4(a, ld2(B1 + k), c1);
    c2 = wmma4(a, ld2(B2 + k), c2);
    c3 = wmma4(a, ld2(B3 + k), c3);
  }

  const float cv0 = cv[col0 +      l];
  const float cv1 = cv[col0 + 16 + l];
  const float cv2 = cv[col0 + 32 + l];
  const float cv3 = cv[col0 + 48 + l];
  #pragma unroll
  for (int r = 0; r < 8; ++r) {
    const int t = row0 + r + 8 * half;
    const float ctv = ct[t];
    float* o = out + (size_t)t * V + col0;
    o[     l] = c0[r] + cv0 + ctv;
    o[16 + l] = c1[r] + cv1 + ctv;
    o[32 + l] = c2[r] + cv2 + ctv;
    o[48 + l] = c3[r] + cv3 + ctv;
  }
}

// ---------------------------------------------------------------------------
extern "C" void kernel_launch(void* const* d_in, const int* in_sizes, int n_in,
                              void* d_out, int out_size, void* d_ws, size_t ws_size,
                              hipStream_t stream) {
  const float* h    = (const float*)d_in[0];  // [S,B,D] -> [T,D]
  const float* emb  = (const float*)d_in[1];  // [V,D]
  const float* Wx   = (const float*)d_in[2];  // [D,D]
  const float* wt   = (const float*)d_in[3];  // [D]
  const float* bvec = (const float*)d_in[4];  // [D]
  const int T = in_sizes[0] / D_DIM;          // 2048
  const int V = in_sizes[1] / D_DIM;          // 32000
  float* out = (float*)d_out;                 // [T,V]
  float* cv  = (float*)d_ws;                  // [V]
  float* ct  = cv + V;                        // [T]

  hsq_kernel<<<(T + 255) / 256, 256, 0, stream>>>(h, ct, T);
  cnf_kernel<<<V / 16, 256, 0, stream>>>(emb, Wx, wt, bvec, cv, V);
  logp_kernel<<<dim3(V / 64, T / 16), 32, 0, stream>>>(h, emb, cv, ct, out, V);
}